// STLSTM_36189394437046
// MI455X (gfx1250) — compile-verified
//
#include <hip/hip_runtime.h>

typedef float v2f __attribute__((ext_vector_type(2)));
typedef float v8f __attribute__((ext_vector_type(8)));

namespace {
constexpr int kB = 64, kT = 256, kE = 512, kH = 512;
constexpr int kM  = kB * kT;   // 16384 rows (time-parallel GEMMs)
constexpr int kH3 = 3 * kH;    // 1536
constexpr int kH4 = 4 * kH;    // 2048
constexpr int LDS_STRIDE  = 144;  // 128 cols + pad (bank-conflict dodge)
constexpr int SCAN_BLOCKS = 16;
}

__device__ __forceinline__ v8f wmma_f32(v2f a, v2f b, v8f c) {
  // D = A(16x4,f32) * B(4x16,f32) + C(16x16,f32)  -> v_wmma_f32_16x16x4_f32
  return __builtin_amdgcn_wmma_f32_16x16x4_f32(false, a, false, b, (short)0, c,
                                               false, false);
}

__device__ __forceinline__ float sigmoidf_(float x) {
  return 1.0f / (1.0f + __expf(-x));
}

// ---------------------------------------------------------------- init
__global__ void stlstm_init(float* __restrict__ h0, unsigned* __restrict__ bar) {
  int i = blockIdx.x * blockDim.x + threadIdx.x;
  if (i < kB * kH) h0[i] = 0.0f;
  if (i == 0) *bar = 0u;
}

// ------------------------------------------------- xp = gather(loc_emb)@Wi + bi
// One wave computes a 16x64 output tile (4 accumulators along N).
__global__ void stlstm_xp_gemm(const int* __restrict__ loc,
                               const float* __restrict__ loc_emb,
                               const float* __restrict__ Wi,
                               const float* __restrict__ bi,
                               float* __restrict__ xp) {
  const int wid  = (blockIdx.x * blockDim.x + threadIdx.x) >> 5;
  const int lane = threadIdx.x & 31;
  const int half = lane >> 4;
  const int l16  = lane & 15;
  const int ntile4 = kH4 / 64;        // 32
  const int mt = wid / ntile4;
  const int nt = wid % ntile4;
  const int row = mt * 16 + l16;
  const float* arow = loc_emb + (size_t)loc[row] * kE;  // fused gather

  v8f acc[4] = {};
  for (int k = 0; k < kE; k += 4) {
    const int kk = k + 2 * half;
    v2f a;
    a.x = arow[kk];
    a.y = arow[kk + 1];
    const float* bk0 = Wi + (size_t)kk * kH4;
    const float* bk1 = bk0 + kH4;
#pragma unroll
    for (int nn = 0; nn < 4; ++nn) {
      const int col = nt * 64 + nn * 16 + l16;
      v2f b;
      b.x = bk0[col];
      b.y = bk1[col];
      acc[nn] = wmma_f32(a, b, acc[nn]);
    }
  }
#pragma unroll
  for (int nn = 0; nn < 4; ++nn) {
    const int col = nt * 64 + nn * 16 + l16;
    const float bv = bi[col];
#pragma unroll
    for (int r = 0; r < 8; ++r) {
      const int orow = mt * 16 + r + 8 * half;
      xp[(size_t)orow * kH4 + col] = acc[nn][r] + bv;
    }
  }
}

// --------------------------------- Fp = q@Wt + sp@Ws  (concat-K GEMM, K=1024)
__global__ void stlstm_fp_gemm(const int* __restrict__ tu, const int* __restrict__ tl,
                               const int* __restrict__ su, const int* __restrict__ sl,
                               const float* __restrict__ tue, const float* __restrict__ tle,
                               const float* __restrict__ sue, const float* __restrict__ sle,
                               const float* __restrict__ Wt, const float* __restrict__ Ws,
                               float* __restrict__ fp) {
  const int wid  = (blockIdx.x * blockDim.x + threadIdx.x) >> 5;
  const int lane = threadIdx.x & 31;
  const int half = lane >> 4;
  const int l16  = lane & 15;
  const int ntile4 = kH3 / 64;        // 24
  const int mt = wid / ntile4;
  const int nt = wid % ntile4;
  const int row  = mt * 16 + l16;
  const int trow = row % kT;          // time index; first step is masked to 0
  const float tsc = (trow > 0) ? (1.0f / 3600.0f) : 0.0f;
  const float ssc = (trow > 0) ? (1.0f / 1000.0f) : 0.0f;
  const float* tur = tue + (size_t)tu[row] * kE;
  const float* tlr = tle + (size_t)tl[row] * kE;
  const float* sur = sue + (size_t)su[row] * kE;
  const float* slr = sle + (size_t)sl[row] * kE;

  v8f acc[4] = {};
  // K half 1: q @ Wt  (gather + add + scale fused into A fragment)
  for (int k = 0; k < kE; k += 4) {
    const int kk = k + 2 * half;
    v2f a;
    a.x = (tur[kk] + tlr[kk]) * tsc;
    a.y = (tur[kk + 1] + tlr[kk + 1]) * tsc;
    const float* bk0 = Wt + (size_t)kk * kH3;
    const float* bk1 = bk0 + kH3;
#pragma unroll
    for (int nn = 0; nn < 4; ++nn) {
      const int col = nt * 64 + nn * 16 + l16;
      v2f b;
      b.x = bk0[col];
      b.y = bk1[col];
      acc[nn] = wmma_f32(a, b, acc[nn]);
    }
  }
  // K half 2: sp @ Ws
  for (int k = 0; k < kE; k += 4) {
    const int kk = k + 2 * half;
    v2f a;
    a.x = (sur[kk] + slr[kk]) * ssc;
    a.y = (sur[kk + 1] + slr[kk + 1]) * ssc;
    const float* bk0 = Ws + (size_t)kk * kH3;
    const float* bk1 = bk0 + kH3;
#pragma unroll
    for (int nn = 0; nn < 4; ++nn) {
      const int col = nt * 64 + nn * 16 + l16;
      v2f b;
      b.x = bk0[col];
      b.y = bk1[col];
      acc[nn] = wmma_f32(a, b, acc[nn]);
    }
  }
#pragma unroll
  for (int nn = 0; nn < 4; ++nn) {
    const int col = nt * 64 + nn * 16 + l16;
#pragma unroll
    for (int r = 0; r < 8; ++r) {
      const int orow = mt * 16 + r + 8 * half;
      fp[(size_t)orow * kH3 + col] = acc[nn][r];
    }
  }
}

// ------------------------------------------------- persistent LSTM scan kernel
// 16 blocks x 256 threads. Each block caches a 512x128 Wh slice in 288KB LDS.
// Each wave owns a 16(batch) x 16(hidden) tile; c-state lives in 8 VGPRs
// across all 256 steps. Device-wide atomic barrier between steps.
__global__ void __launch_bounds__(256, 1)
stlstm_scan(const float* __restrict__ xp, const float* __restrict__ fp,
            const float* __restrict__ Wh,
            float* __restrict__ h0, float* __restrict__ h1,
            float* __restrict__ all_h, float* __restrict__ all_c,
            unsigned* __restrict__ bar) {
  extern __shared__ float smem[];
  const int lane = threadIdx.x & 31;
  const int wid  = threadIdx.x >> 5;      // 0..7
  const int half = lane >> 4;
  const int l16  = lane & 15;
  const int mt = wid & 3;                 // batch tile 0..3
  const int jl = wid >> 2;                // local hidden tile 0..1
  const int jt = blockIdx.x * 2 + jl;     // global hidden tile 0..31

  // Stage Wh columns this block needs: 4 gates x 2 tiles x 16 cols = 128 cols.
  for (int i = threadIdx.x; i < 512 * 128; i += 256) {
    const int k   = i >> 7;
    const int lc  = i & 127;
    const int g   = lc >> 5;
    const int rem = lc & 31;
    const int gcol = g * kH + (blockIdx.x * 2 + (rem >> 4)) * 16 + (rem & 15);
    smem[k * LDS_STRIDE + lc] = Wh[(size_t)k * kH4 + gcol];
  }
  __syncthreads();

  float creg[8] = {0.f, 0.f, 0.f, 0.f, 0.f, 0.f, 0.f, 0.f};

  for (int t = 0; t < kT; ++t) {
    const float* hprev = (t & 1) ? h1 : h0;
    float* hnext       = (t & 1) ? h0 : h1;
    const float* arow = hprev + (size_t)(mt * 16 + l16) * kH;

    v8f acc[4] = {};
    for (int k = 0; k < kH; k += 4) {
      const int kk = k + 2 * half;
      v2f a;
      a.x = arow[kk];
      a.y = arow[kk + 1];
      const float* s0 = smem + kk * LDS_STRIDE;
      const float* s1 = s0 + LDS_STRIDE;
#pragma unroll
      for (int g = 0; g < 4; ++g) {
        const int lc = g * 32 + jl * 16 + l16;
        v2f b;
        b.x = s0[lc];
        b.y = s1[lc];
        acc[g] = wmma_f32(a, b, acc[g]);
      }
    }

    const int hcol = jt * 16 + l16;
#pragma unroll
    for (int r = 0; r < 8; ++r) {
      const int brow = mt * 16 + r + 8 * half;
      const size_t bt = (size_t)brow * kT + t;
      const float* xpr = xp + bt * kH4;
      const float* fpr = fp + bt * kH3;
      const float ti = acc[0][r] + xpr[hcol]          + fpr[hcol];
      const float tf = acc[1][r] + xpr[kH + hcol]     + fpr[kH + hcol];
      const float to = acc[2][r] + xpr[2 * kH + hcol] + fpr[2 * kH + hcol];
      const float tg = acc[3][r] + xpr[3 * kH + hcol];
      const float iv = sigmoidf_(ti);
      const float fv = sigmoidf_(tf);
      const float ov = sigmoidf_(to);
      const float gv = tanhf(tg);
      const float cv = fv * creg[r] + iv * gv;
      creg[r] = cv;
      const float hv = ov * tanhf(cv);
      hnext[(size_t)brow * kH + hcol] = hv;
      all_h[bt * kH + hcol] = hv;
      all_c[bt * kH + hcol] = cv;
    }

    // device-wide step barrier (release/acquire)
    __threadfence();
    __syncthreads();
    if (threadIdx.x == 0) {
      atomicAdd(bar, 1u);
      const unsigned target = (unsigned)(t + 1) * SCAN_BLOCKS;
      while (__hip_atomic_load(bar, __ATOMIC_ACQUIRE,
                               __HIP_MEMORY_SCOPE_AGENT) < target) {
        __builtin_amdgcn_s_sleep(1);
      }
    }
    __syncthreads();
    __threadfence();
  }
}

// ---------------------------------------------------------------- tail gather
__global__ void stlstm_tail(const int* __restrict__ vlen,
                            const float* __restrict__ all_h,
                            const float* __restrict__ all_c,
                            float* __restrict__ last_h,
                            float* __restrict__ last_c) {
  const int i = blockIdx.x * blockDim.x + threadIdx.x;
  if (i >= kB * kH) return;
  const int b = i / kH, j = i % kH;
  int v = vlen[b];                       // in [1, T]
  int tsel = (v <= 0) ? (kT - 1) : (v - 1);   // index 0 selects hT == all_h[T-1]
  if (tsel >= kT) tsel = kT - 1;
  const size_t src = ((size_t)b * kT + tsel) * kH + j;
  last_h[i] = all_h[src];
  last_c[i] = all_c[src];
}

extern "C" void kernel_launch(void* const* d_in, const int* in_sizes, int n_in,
                              void* d_out, int out_size, void* d_ws, size_t ws_size,
                              hipStream_t stream) {
  (void)in_sizes; (void)n_in; (void)out_size; (void)ws_size;
  const int*   loc     = (const int*)d_in[0];
  const int*   tu      = (const int*)d_in[1];
  const int*   tl      = (const int*)d_in[2];
  const int*   su      = (const int*)d_in[3];
  const int*   sl      = (const int*)d_in[4];
  const int*   vlen    = (const int*)d_in[5];
  const float* loc_emb = (const float*)d_in[6];
  const float* tue     = (const float*)d_in[7];
  const float* tle     = (const float*)d_in[8];
  const float* sue     = (const float*)d_in[9];
  const float* sle     = (const float*)d_in[10];
  const float* Wt      = (const float*)d_in[11];
  const float* Ws      = (const float*)d_in[12];
  const float* Wi      = (const float*)d_in[13];
  const float* bi      = (const float*)d_in[14];
  const float* Wh      = (const float*)d_in[15];

  float* out    = (float*)d_out;
  float* all_h  = out;                                   // (B,T,H)
  float* last_h = out + (size_t)kB * kT * kH;            // (B,H)
  float* last_c = last_h + (size_t)kB * kH;              // (B,H)

  float* ws = (float*)d_ws;
  size_t off = 0;
  float* xp    = ws + off; off += (size_t)kM * kH4;      // 33.55M f
  float* fp    = ws + off; off += (size_t)kM * kH3;      // 25.17M f
  float* all_c = ws + off; off += (size_t)kM * kH;       //  8.39M f
  float* h0    = ws + off; off += (size_t)kB * kH;
  float* h1    = ws + off; off += (size_t)kB * kH;
  unsigned* bar = (unsigned*)(ws + off);

  stlstm_init<<<(kB * kH + 255) / 256, 256, 0, stream>>>(h0, bar);

  {
    const int waves = (kM / 16) * (kH4 / 64);            // 32768
    stlstm_xp_gemm<<<waves * 32 / 256, 256, 0, stream>>>(loc, loc_emb, Wi, bi, xp);
  }
  {
    const int waves = (kM / 16) * (kH3 / 64);            // 24576
    stlstm_fp_gemm<<<waves * 32 / 256, 256, 0, stream>>>(tu, tl, su, sl,
                                                         tue, tle, sue, sle,
                                                         Wt, Ws, fp);
  }

  const size_t lds_bytes = (size_t)512 * LDS_STRIDE * sizeof(float);  // 288 KB
  stlstm_scan<<<SCAN_BLOCKS, 256, lds_bytes, stream>>>(xp, fp, Wh, h0, h1,
                                                       all_h, all_c, bar);

  stlstm_tail<<<(kB * kH + 255) / 256, 256, 0, stream>>>(vlen, all_h, all_c,
                                                         last_h, last_c);
}